// SimpleSAGE_28020366639688
// MI455X (gfx1250) — compile-verified
//
#include <hip/hip_runtime.h>

typedef float v2f __attribute__((ext_vector_type(2)));
typedef float v8f __attribute__((ext_vector_type(8)));

#define N_NODES 50000
#define N_EDGES 800000
#define NUM_GRAPHS 500
#define NODES_PER_GRAPH 100
#define IN_CH 128
#define HID 64
#define G_PAD 512   // 32 row tiles of 16 for layer-2 GEMM

// ---------------------------------------------------------------- zero
__global__ void zero_f32(float* __restrict__ p, int n) {
    int i = blockIdx.x * blockDim.x + threadIdx.x;
    int stride = gridDim.x * blockDim.x;
    for (; i < n; i += stride) p[i] = 0.0f;
}

// ---------------------------------------------------------------- layer-1 scatter
// one wave32 per edge; lane handles 4 channels (float4 load, 4 atomic adds)
__global__ void scatter1(const float* __restrict__ x,
                         const int* __restrict__ src, const int* __restrict__ dst,
                         float* __restrict__ agg, float* __restrict__ deg) {
    int t = blockIdx.x * blockDim.x + threadIdx.x;
    int e = t >> 5;
    int lane = t & 31;
    if (e >= N_EDGES) return;
    int s = src[e];
    int d = dst[e];
    const float4* xs = (const float4*)(x + (size_t)s * IN_CH);
    float4 v = xs[lane];                       // channels lane*4 .. lane*4+3
    float* a = agg + (size_t)d * IN_CH + lane * 4;
    atomicAdd(a + 0, v.x);
    atomicAdd(a + 1, v.y);
    atomicAdd(a + 2, v.z);
    atomicAdd(a + 3, v.w);
    if (lane == 0) atomicAdd(deg + d, 1.0f);
}

// ---------------------------------------------------------------- layer-1 GEMM (WMMA f32 16x16x4)
// block = 128 threads = 4 waves; block handles 16 rows, wave w handles cols [16w,16w+16)
// h1 = relu(mean @ Wl^T + x @ Wr^T + b)
__global__ void gemm1(const float* __restrict__ x, const float* __restrict__ agg,
                      const float* __restrict__ deg,
                      const float* __restrict__ Wl, const float* __restrict__ Wr,
                      const float* __restrict__ b1, float* __restrict__ h1) {
    __shared__ float xt[16][IN_CH];
    __shared__ float mt[16][IN_CH];
    __shared__ float rdeg[16];

    const int tid = threadIdx.x;
    const int m0 = blockIdx.x * 16;

    if (tid < 16) rdeg[tid] = 1.0f / fmaxf(deg[m0 + tid], 1.0f);
    __syncthreads();

    for (int i = tid; i < 16 * IN_CH; i += 128) {
        int r = i >> 7;          // / 128
        int c = i & (IN_CH - 1); // % 128
        size_t off = (size_t)(m0 + r) * IN_CH + c;
        xt[r][c] = x[off];
        mt[r][c] = agg[off] * rdeg[r];
    }
    __syncthreads();

    const int wave = tid >> 5;
    const int lane = tid & 31;
    const int n0 = wave * 16;
    const int ml = lane & 15;            // A-row / B-col / D-col for this lane
    const int koff = (lane >> 4) << 1;   // K offset: lanes 16-31 hold K=2,3

    v8f acc = {};
    for (int k = 0; k < IN_CH; k += 4) {
        v2f am = *(const v2f*)&mt[ml][k + koff];
        v2f ax = *(const v2f*)&xt[ml][k + koff];
        v2f bl = *(const v2f*)(Wl + (size_t)(n0 + ml) * IN_CH + k + koff);
        v2f br = *(const v2f*)(Wr + (size_t)(n0 + ml) * IN_CH + k + koff);
        acc = __builtin_amdgcn_wmma_f32_16x16x4_f32(false, am, false, bl,
                                                    (short)0, acc, false, false);
        acc = __builtin_amdgcn_wmma_f32_16x16x4_f32(false, ax, false, br,
                                                    (short)0, acc, false, false);
    }

    const float bias = b1[n0 + ml];
    const int mhi = (lane >> 4) << 3;    // 0 or 8
#pragma unroll
    for (int r = 0; r < 8; ++r) {
        int m = r + mhi;
        float v = acc[r] + bias;
        h1[(size_t)(m0 + m) * HID + n0 + ml] = fmaxf(v, 0.0f);
    }
}

// ---------------------------------------------------------------- layer-2 scatter (center nodes only)
__global__ void scatter2(const float* __restrict__ h1,
                         const int* __restrict__ src, const int* __restrict__ dst,
                         const int* __restrict__ cpos,
                         float* __restrict__ agg2, float* __restrict__ deg2) {
    int e = blockIdx.x * blockDim.x + threadIdx.x;
    if (e >= N_EDGES) return;
    int d = dst[e];
    int g = d / NODES_PER_GRAPH;
    int pos = d - g * NODES_PER_GRAPH;
    if (cpos[g] != pos) return;          // dst is not the center node of its graph
    int s = src[e];
    const float* hs = h1 + (size_t)s * HID;
    float* a = agg2 + (size_t)g * HID;
#pragma unroll 4
    for (int c = 0; c < HID; ++c) atomicAdd(a + c, hs[c]);
    atomicAdd(deg2 + g, 1.0f);
}

// ---------------------------------------------------------------- layer-2 GEMM (512 padded rows)
__global__ void gemm2(const float* __restrict__ h1, const float* __restrict__ agg2,
                      const float* __restrict__ deg2, const int* __restrict__ cpos,
                      const float* __restrict__ Wl, const float* __restrict__ Wr,
                      const float* __restrict__ b2, float* __restrict__ h2) {
    __shared__ float ht[16][HID];
    __shared__ float mt[16][HID];
    __shared__ float rdeg[16];

    const int tid = threadIdx.x;
    const int m0 = blockIdx.x * 16;

    if (tid < 16) rdeg[tid] = 1.0f / fmaxf(deg2[m0 + tid], 1.0f);
    __syncthreads();

    for (int i = tid; i < 16 * HID; i += 128) {
        int r = i >> 6;          // / 64
        int c = i & (HID - 1);   // % 64
        int g = m0 + r;
        int cg = (g < NUM_GRAPHS) ? (g * NODES_PER_GRAPH + cpos[g]) : 0;
        ht[r][c] = h1[(size_t)cg * HID + c];
        mt[r][c] = agg2[(size_t)g * HID + c] * rdeg[r];  // rows >= 500 are zeroed ws
    }
    __syncthreads();

    const int wave = tid >> 5;
    const int lane = tid & 31;
    const int n0 = wave * 16;
    const int ml = lane & 15;
    const int koff = (lane >> 4) << 1;

    v8f acc = {};
    for (int k = 0; k < HID; k += 4) {
        v2f am = *(const v2f*)&mt[ml][k + koff];
        v2f ax = *(const v2f*)&ht[ml][k + koff];
        v2f bl = *(const v2f*)(Wl + (size_t)(n0 + ml) * HID + k + koff);
        v2f br = *(const v2f*)(Wr + (size_t)(n0 + ml) * HID + k + koff);
        acc = __builtin_amdgcn_wmma_f32_16x16x4_f32(false, am, false, bl,
                                                    (short)0, acc, false, false);
        acc = __builtin_amdgcn_wmma_f32_16x16x4_f32(false, ax, false, br,
                                                    (short)0, acc, false, false);
    }

    const float bias = b2[n0 + ml];
    const int mhi = (lane >> 4) << 3;
#pragma unroll
    for (int r = 0; r < 8; ++r) {
        int g = m0 + r + mhi;
        if (g < NUM_GRAPHS) {
            float v = acc[r] + bias;
            h2[(size_t)g * HID + n0 + ml] = fmaxf(v, 0.0f);
        }
    }
}

// ---------------------------------------------------------------- readout
__global__ void readout(const float* __restrict__ h2, const float* __restrict__ Wout,
                        const float* __restrict__ bout, float* __restrict__ out) {
    int g = blockIdx.x * blockDim.x + threadIdx.x;
    if (g >= NUM_GRAPHS) return;
    float s = bout[0];
    const float* h = h2 + (size_t)g * HID;
#pragma unroll 8
    for (int c = 0; c < HID; ++c) s += Wout[c] * h[c];
    out[g] = s;
}

// ---------------------------------------------------------------- launch
extern "C" void kernel_launch(void* const* d_in, const int* in_sizes, int n_in,
                              void* d_out, int out_size, void* d_ws, size_t ws_size,
                              hipStream_t stream) {
    const float* x    = (const float*)d_in[0];
    const int*   ei   = (const int*)d_in[1];
    /* d_in[2] = batch: unused (batch == arange // 100 by construction) */
    const int*   cpos = (const int*)d_in[3];
    const float* Wl1  = (const float*)d_in[4];
    const float* Wr1  = (const float*)d_in[5];
    const float* b1   = (const float*)d_in[6];
    const float* Wl2  = (const float*)d_in[7];
    const float* Wr2  = (const float*)d_in[8];
    const float* b2   = (const float*)d_in[9];
    const float* Wout = (const float*)d_in[10];
    const float* bout = (const float*)d_in[11];

    const int* src = ei;
    const int* dst = ei + N_EDGES;

    float* ws   = (float*)d_ws;
    float* agg1 = ws;                               // N_NODES*IN_CH
    float* deg1 = agg1 + (size_t)N_NODES * IN_CH;   // N_NODES
    float* h1   = deg1 + N_NODES;                   // N_NODES*HID
    float* agg2 = h1 + (size_t)N_NODES * HID;       // G_PAD*HID
    float* deg2 = agg2 + (size_t)G_PAD * HID;       // G_PAD
    float* h2   = deg2 + G_PAD;                     // NUM_GRAPHS*HID
    float* out  = (float*)d_out;

    zero_f32<<<2048, 256, 0, stream>>>(agg1, N_NODES * IN_CH + N_NODES); // agg1+deg1 contiguous
    zero_f32<<<64,   256, 0, stream>>>(agg2, G_PAD * HID + G_PAD);       // agg2+deg2 contiguous

    scatter1<<<(N_EDGES * 32 + 255) / 256, 256, 0, stream>>>(x, src, dst, agg1, deg1);
    gemm1<<<N_NODES / 16, 128, 0, stream>>>(x, agg1, deg1, Wl1, Wr1, b1, h1);

    scatter2<<<(N_EDGES + 255) / 256, 256, 0, stream>>>(h1, src, dst, cpos, agg2, deg2);
    gemm2<<<G_PAD / 16, 128, 0, stream>>>(h1, agg2, deg2, cpos, Wl2, Wr2, b2, h2);

    readout<<<(NUM_GRAPHS + 127) / 128, 128, 0, stream>>>(h2, Wout, bout, out);
}